// GCNLayer_83382495084579
// MI455X (gfx1250) — compile-verified
//
#include <hip/hip_runtime.h>
#include <hip/hip_bf16.h>

#define IN_F  32
#define OUT_F 32

typedef float v2f __attribute__((ext_vector_type(2)));
typedef float v8f __attribute__((ext_vector_type(8)));

// ---------------------------------------------------------------------------
// Kernel 0: zero the node-accumulator workspace (graph-capture safe memset).
// ---------------------------------------------------------------------------
__global__ void gcn_zero_ws(float* __restrict__ ws, int n) {
    int i = blockIdx.x * blockDim.x + threadIdx.x;
    int stride = gridDim.x * blockDim.x;
    for (; i < n; i += stride) ws[i] = 0.0f;
}

// ---------------------------------------------------------------------------
// Kernel 1: edge gather + scatter-sum.
//   8 lanes per edge; each lane moves a float4 chunk of the 32-float feature
//   row (8 x 16B = one 128B row) and atomically accumulates into h[dst].
//   feature (12.8 MB) and h (12.8 MB) both fit in the 192 MB L2, so this is
//   L2-bandwidth/atomic-throughput bound -- keep it maximally simple & wide.
// ---------------------------------------------------------------------------
__global__ void gcn_scatter_sum(const float* __restrict__ feat,
                                const int*   __restrict__ src,
                                const int*   __restrict__ dst,
                                float*       __restrict__ h,
                                int n_edges) {
    long gid = (long)blockIdx.x * blockDim.x + threadIdx.x;
    long e   = gid >> 3;              // 8 lanes per edge
    if (e >= n_edges) return;
    int  c   = ((int)gid & 7) * 4;    // feature chunk [c, c+4)

    int s = src[e];
    int d = dst[e];

    const float4 v = *(const float4*)(feat + (long)s * IN_F + c);
    float* o = h + (long)d * IN_F + c;
    atomicAdd(o + 0, v.x);
    atomicAdd(o + 1, v.y);
    atomicAdd(o + 2, v.z);
    atomicAdd(o + 3, v.w);
}

// ---------------------------------------------------------------------------
// Kernel 2: out = h @ W^T + b via V_WMMA_F32_16X16X4_F32 (full f32 precision).
//   One wave computes a 16x16 output tile; K=32 as 8 chained K=4 WMMAs.
//   blockDim = 64 -> wave0 handles cols [0,16), wave1 handles cols [16,32).
//
//   f32 16x16x4 A layout (2 VGPRs): lane l<16 holds A[l][k], A[l][k+1];
//   lane l>=16 holds A[l-16][k+2], A[l-16][k+3]  -> one float2 load at
//   row*32 + k + (lane>>4)*2.  B[k][n] = W[n][k], identical addressing with
//   the W row selected by the output column.  C/D f32 16x16: VGPR r, lane
//   n<16 -> (row r, col n); lane n>=16 -> (row r+8, col n-16).
//
//   Stores: the full-tile case is a *uniform* condition, so take a scalar
//   branch into an unguarded path -> one base address + 8 stores with
//   immediate offsets (r * OUT_F * 4 bytes), no per-lane exec churn.
// ---------------------------------------------------------------------------
__global__ void gcn_gemm_wmma(const float* __restrict__ h,
                              const float* __restrict__ W,
                              const float* __restrict__ b,
                              float*       __restrict__ out,
                              int n_nodes) {
    const int lane = threadIdx.x & 31;
    const int n0   = (threadIdx.x >> 5) * 16;   // output-col tile (0 or 16)
    const int m0   = blockIdx.x * 16;           // output-row tile
    const int lm   = lane & 15;
    const int hi   = lane >> 4;                 // 0: K pair {k,k+1}, 1: {k+2,k+3}

    // Seed accumulator with bias: every C VGPR, col n0+lm.
    const float bias = b[n0 + lm];
    v8f c;
#pragma unroll
    for (int r = 0; r < 8; ++r) c[r] = bias;

    int arow = m0 + lm;
    if (arow >= n_nodes) arow = n_nodes - 1;    // clamp: keep EXEC all-1s
    const float* aptr = h + (long)arow * IN_F + hi * 2;
    const float* bptr = W + (long)(n0 + lm) * IN_F + hi * 2;

#pragma unroll
    for (int k = 0; k < IN_F; k += 4) {
        v2f a  = *(const v2f*)(aptr + k);
        v2f bm = *(const v2f*)(bptr + k);
        // 8 args: (neg_a, A, neg_b, B, c_mod, C, reuse_a, reuse_b)
        c = __builtin_amdgcn_wmma_f32_16x16x4_f32(
                false, a, false, bm, (short)0, c, false, false);
    }

    if (m0 + 16 <= n_nodes) {
        // Fast path (uniform): base + immediate offsets, no guards.
        float* obase = out + (long)(m0 + hi * 8) * OUT_F + n0 + lm;
#pragma unroll
        for (int r = 0; r < 8; ++r)
            obase[r * OUT_F] = c[r];
    } else {
        // Ragged tail (never taken for N % 16 == 0, kept for generality).
#pragma unroll
        for (int r = 0; r < 8; ++r) {
            int row = m0 + r + hi * 8;
            if (row < n_nodes)
                out[(long)row * OUT_F + n0 + lm] = c[r];
        }
    }
}

// ---------------------------------------------------------------------------
extern "C" void kernel_launch(void* const* d_in, const int* in_sizes, int n_in,
                              void* d_out, int out_size, void* d_ws, size_t ws_size,
                              hipStream_t stream) {
    const float* feat = (const float*)d_in[0];   // [N, 32] f32
    const int*   src  = (const int*)  d_in[1];   // [E] i32
    const int*   dst  = (const int*)  d_in[2];   // [E] i32
    const float* W    = (const float*)d_in[3];   // [32, 32] f32
    const float* b    = (const float*)d_in[4];   // [32] f32
    float*       out  = (float*)d_out;           // [N, 32] f32
    float*       h    = (float*)d_ws;            // [N, 32] f32 accumulator

    const int n_nodes = in_sizes[0] / IN_F;
    const int n_edges = in_sizes[1];
    const int hsz     = n_nodes * IN_F;

    // 1) zero accumulator
    {
        int blocks = (hsz + 1023) / 1024;
        if (blocks > 4096) blocks = 4096;        // grid-stride
        gcn_zero_ws<<<blocks, 256, 0, stream>>>(h, hsz);
    }

    // 2) gather + scatter-sum over edges (8 lanes per edge)
    {
        long threads = (long)n_edges * 8;
        int  blocks  = (int)((threads + 255) / 256);
        gcn_scatter_sum<<<blocks, 256, 0, stream>>>(feat, src, dst, h, n_edges);
    }

    // 3) dense layer via f32 WMMA: 2 waves/block cover the 32 output cols
    {
        int blocks = (n_nodes + 15) / 16;
        gcn_gemm_wmma<<<blocks, 64, 0, stream>>>(h, W, b, out, n_nodes);
    }
}